// post_process_85126251807580
// MI455X (gfx1250) — compile-verified
//
#include <hip/hip_runtime.h>
#include <stdint.h>

#define WINDOW   30
#define ROWS     2048
#define ROWLEN   16384
#define NWIN     (ROWLEN - WINDOW + 1)   // 16355 valid window starts
#define TILE     2048                    // elements (and window starts) per tile
#define NTILES   (ROWLEN / TILE)         // 8
#define TPB      256                     // 8 wave32s
#define EPT      (TILE / TPB)            // 8 elements / windows per thread

// ---------------------------------------------------------------------------
// CDNA5 async global->LDS copy (GLOBAL_LOAD_ASYNC_TO_LDS_B128, ASYNCcnt).
// GVS addressing: mem = SGPR64 + VGPR32_offset;  LDS dest addr from VGPR.
// Low 32 bits of a generic pointer to a __shared__ object are the LDS byte
// offset (flat shared aperture puts the offset in addr[31:0]).
// ---------------------------------------------------------------------------
__device__ __forceinline__ void async_tile_load(const float* __restrict__ g,
                                                const float* l, int t) {
    const uint32_t l0 = (uint32_t)(uintptr_t)l;   // LDS byte address of tile
    const uint32_t o0 = (uint32_t)(t * 16);       // this lane's byte offset
    // two b128 per lane: 256 lanes x 2 x 16B = 8192B = 2048 floats
    asm volatile("global_load_async_to_lds_b128 %0, %1, %2"
                 :
                 : "v"(l0 + o0), "v"(o0), "s"((uint64_t)(uintptr_t)g)
                 : "memory");
    asm volatile("global_load_async_to_lds_b128 %0, %1, %2"
                 :
                 : "v"(l0 + o0 + 4096u), "v"(o0 + 4096u), "s"((uint64_t)(uintptr_t)g)
                 : "memory");
}

#define WAIT_ASYNC(n) asm volatile("s_wait_asynccnt %0" ::"i"(n) : "memory")

__device__ __forceinline__ float sigmoid_f32(float v) {
    return 1.0f / (1.0f + __expf(-v));
}

__global__ void __launch_bounds__(TPB)
win_mean_max_kernel(const float* __restrict__ x,
                    const float* __restrict__ mask,
                    float* __restrict__ out) {
    __shared__ __align__(16) float Xs[2][TILE];      // raw x tiles (double buf)
    __shared__ __align__(16) float Ms[2][TILE];      // raw mask tiles
    __shared__ __align__(16) float Ps[TILE + 32];    // prob tile + halo

    const int row = blockIdx.x;
    const int t   = threadIdx.x;
    const float* xr = x    + (size_t)row * ROWLEN;
    const float* mr = mask + (size_t)row * ROWLEN;

    // prefetch tile 0 (ASYNCcnt += 4)
    async_tile_load(xr, Xs[0], t);
    async_tile_load(mr, Ms[0], t);

    float best = -3.402823466e38f;

    for (int i = 0; i < NTILES; ++i) {
        const int cur  = i & 1;
        const int nxt  = cur ^ 1;
        const int base = i * TILE;

        if (i + 1 < NTILES) {
            // issue next tile into the other buffer (safe: that buffer was
            // last read before the 2nd barrier of iteration i-1)
            async_tile_load(xr + base + TILE, Xs[nxt], t);
            async_tile_load(mr + base + TILE, Ms[nxt], t);
            WAIT_ASYNC(4);   // in-order: first 4 (tile i) complete
        } else {
            WAIT_ASYNC(0);   // nothing in flight behind tile i
        }
        __syncthreads();     // all waves' tile-i data visible; prev P reads done

        // --- prob = sigmoid(x)*mask for this thread's 8 elements -----------
        float p[EPT];
        const int a = t * EPT;
#pragma unroll
        for (int j = 0; j < EPT; ++j) {
            p[j] = sigmoid_f32(Xs[cur][a + j]) * Ms[cur][a + j];
            Ps[a + j] = p[j];
        }
        // halo prob [base+TILE, base+TILE+29) — tiny direct loads (L2-hot).
        // Last tile needs no halo: window 16354 ends at element 16383.
        if (t < WINDOW - 1 && i + 1 < NTILES) {
            Ps[TILE + t] = sigmoid_f32(xr[base + TILE + t]) * mr[base + TILE + t];
        }
        __syncthreads();

        // --- rolling window sums: 8 consecutive windows per thread ---------
        float s = 0.0f;
#pragma unroll
        for (int j = 0; j < EPT; ++j) s += p[j];          // own values (regs)
#pragma unroll
        for (int j = EPT; j < WINDOW; ++j) s += Ps[a + j]; // 22 LDS reads
        const int ws = base + a;
        if (ws < NWIN) best = fmaxf(best, s);
#pragma unroll
        for (int k = 1; k < EPT; ++k) {
            s += Ps[a + WINDOW - 1 + k] - p[k - 1];
            if (ws + k < NWIN) best = fmaxf(best, s);
        }
        // next iteration's barrier (after WAIT_ASYNC) separates these P reads
        // from the next tile's P writes
    }

    // --- block max reduction (reuse Ps) ------------------------------------
    __syncthreads();
    Ps[t] = best;
    __syncthreads();
    for (int stride = TPB / 2; stride > 0; stride >>= 1) {
        if (t < stride) Ps[t] = fmaxf(Ps[t], Ps[t + stride]);
        __syncthreads();
    }
    if (t == 0) out[row] = Ps[0] / (float)WINDOW;
}

extern "C" void kernel_launch(void* const* d_in, const int* in_sizes, int n_in,
                              void* d_out, int out_size, void* d_ws, size_t ws_size,
                              hipStream_t stream) {
    (void)in_sizes; (void)n_in; (void)d_ws; (void)ws_size; (void)out_size;
    const float* x    = (const float*)d_in[0];
    const float* mask = (const float*)d_in[1];
    float* out        = (float*)d_out;
    win_mean_max_kernel<<<dim3(ROWS), dim3(TPB), 0, stream>>>(x, mask, out);
}